// E3Hamiltonian_38165079392431
// MI455X (gfx1250) — compile-verified
//
#include <hip/hip_runtime.h>
#include <math.h>
#include <stdint.h>

#define FEAT 133
#define EBLK 64   // edge rows per block (2 waves)
#define BPOFF 1024  // float offset of padded WMMA-B tables inside d_ws

typedef float v2f __attribute__((ext_vector_type(2)));
typedef float v8f __attribute__((ext_vector_type(8)));

// ---------------- optional CDNA5 async global<->LDS path ----------------
#define TRY_ASYNC 1
#if TRY_ASYNC && __has_builtin(__builtin_amdgcn_global_load_async_to_lds_b128) && \
    __has_builtin(__builtin_amdgcn_global_store_async_from_lds_b128) && \
    __has_builtin(__builtin_amdgcn_s_wait_asynccnt)
#define HAVE_ASYNC 1
#else
#define HAVE_ASYNC 0
#endif

#if HAVE_ASYNC
// per compiler diagnostic: params are generic pointers to 128-bit int vectors
typedef int vs4 __attribute__((vector_size(4 * sizeof(int))));
__device__ __forceinline__ void async_g2l_b128(const void* g, void* l) {
  __builtin_amdgcn_global_load_async_to_lds_b128((vs4*)g, (vs4*)l, 0, 0);
}
__device__ __forceinline__ void async_l2g_b128(void* g, const void* l) {
  __builtin_amdgcn_global_store_async_from_lds_b128((vs4*)g, (vs4*)l, 0, 0);
}
#endif

// ======================= CG (Wigner-3j) init kernel ======================
struct cpx { double re, im; };
__device__ __forceinline__ cpx cmul(cpx a, cpx b) {
  cpx r; r.re = a.re*b.re - a.im*b.im; r.im = a.re*b.im + a.im*b.re; return r;
}

__device__ double dfact(int n) { double r = 1.0; for (int i = 2; i <= n; ++i) r *= (double)i; return r; }

__device__ double su2cg_coeff(int j1,int m1,int j2,int m2,int j3,int m3) {
  if (m3 != m1 + m2) return 0.0;
  int vmin = -j1 + j2 + m3;
  if (-j1 + m1 > vmin) vmin = -j1 + m1;
  if (0 > vmin) vmin = 0;
  int vmax = j2 + j3 + m1;
  if (j3 - j1 + j2 < vmax) vmax = j3 - j1 + j2;
  if (j3 + m3 < vmax) vmax = j3 + m3;
  double C = sqrt((2.0*j3 + 1.0) * dfact(j3+j1-j2) * dfact(j3-j1+j2) * dfact(j1+j2-j3) *
                  dfact(j3+m3) * dfact(j3-m3) /
                  (dfact(j1+j2+j3+1) * dfact(j1-m1) * dfact(j1+m1) * dfact(j2-m2) * dfact(j2+m2)));
  double S = 0.0;
  for (int v = vmin; v <= vmax; ++v) {
    double sg = ((v + j2 + m2) & 1) ? -1.0 : 1.0;
    S += sg * dfact(j2+j3+m1-v) * dfact(j1-m1+v) /
         (dfact(v) * dfact(j3-j1+j2-v) * dfact(j3+m3-v) * dfact(v+j1-j2-m3));
  }
  return C * S;
}

__device__ void q_r2c(int l, cpx q[9][9]) {
  int d = 2*l + 1;
  for (int i = 0; i < d; ++i)
    for (int j = 0; j < d; ++j) { q[i][j].re = 0.0; q[i][j].im = 0.0; }
  const double is2 = 0.70710678118654752440;
  for (int m = -l; m < 0; ++m) { q[l+m][l-m].re = is2; q[l+m][l+m].im = -is2; }
  q[l][l].re = 1.0;
  for (int m = 1; m <= l; ++m) {
    double s = (m & 1) ? -1.0 : 1.0;
    q[l+m][l+m].re = s*is2; q[l+m][l-m].im = s*is2;
  }
  cpx ph;
  switch (l & 3) {
    case 0:  ph.re = 1.0;  ph.im = 0.0;  break;
    case 1:  ph.re = 0.0;  ph.im = -1.0; break;
    case 2:  ph.re = -1.0; ph.im = 0.0;  break;
    default: ph.re = 0.0;  ph.im = 1.0;  break;
  }
  for (int i = 0; i < d; ++i)
    for (int j = 0; j < d; ++j) q[i][j] = cmul(ph, q[i][j]);
}

// Phase 1: 14 (pairtype, l3) combos; one thread each.
// CG layout per pairtype: [(l*d2+m)*nrme + r], packed at cgws[0..965].
// Phase 2: build zero-padded WMMA-B tables at cgws[BPOFF..BPOFF+1535]:
//   Bp[kpair][col][2] with value (k<nrme && col<nrme) ? cg[col*nrme+k] : 0
__global__ void init_cg_kernel(float* __restrict__ cgws) {
  const int t = threadIdx.x;
  const int L1[14]   = {0,0,0,1,1,1,1,1,1,2,2,2,2,2};
  const int L2[14]   = {0,1,2,1,1,1,2,2,2,2,2,2,2,2};
  const int L3[14]   = {0,1,2,0,1,2,1,2,3,0,1,2,3,4};
  const int BASE[14] = {0,1,10,35,35,35,116,116,116,341,341,341,341,341};
  const int NR[14]   = {1,3,5,9,9,9,15,15,15,25,25,25,25,25};
  const int ROFF[14] = {0,0,0,0,1,4,0,3,8,0,1,4,9,16};
  if (t < 14) {
    const int l1 = L1[t], l2 = L2[t], l3 = L3[t];
    const int d1 = 2*l1+1, d2 = 2*l2+1, d3 = 2*l3+1;

    double C[5][5][9];
    for (int i = 0; i < d1; ++i)
      for (int k = 0; k < d2; ++k)
        for (int n = 0; n < d3; ++n) C[i][k][n] = 0.0;
    for (int m1 = -l1; m1 <= l1; ++m1)
      for (int m2 = -l2; m2 <= l2; ++m2) {
        int m3 = m1 + m2;
        if (m3 >= -l3 && m3 <= l3)
          C[l1+m1][l2+m2][l3+m3] = su2cg_coeff(l1, m1, l2, m2, l3, m3);
      }
    cpx Q1[9][9], Q2[9][9], Q3[9][9];
    q_r2c(l1, Q1); q_r2c(l2, Q2); q_r2c(l3, Q3);
    double R[5][5][9];
    double n2 = 0.0;
    for (int j = 0; j < d1; ++j)
      for (int lq = 0; lq < d2; ++lq)
        for (int m = 0; m < d3; ++m) {
          double ar = 0.0;
          for (int i = 0; i < d1; ++i)
            for (int k = 0; k < d2; ++k)
              for (int n = 0; n < d3; ++n) {
                double cv = C[i][k][n];
                if (cv == 0.0) continue;
                cpx t1 = cmul(Q1[i][j], Q2[k][lq]);
                cpx q3c; q3c.re = Q3[n][m].re; q3c.im = -Q3[n][m].im;
                cpx t2 = cmul(t1, q3c);
                ar += t2.re * cv;
              }
          R[j][lq][m] = ar;
          n2 += ar * ar;
        }
    double sc = sqrt((2.0*l3 + 1.0) / n2);
    float* dst = cgws + BASE[t];
    const int nr = NR[t], ro = ROFF[t];
    for (int j = 0; j < d1; ++j)
      for (int lq = 0; lq < d2; ++lq)
        for (int m = 0; m < d3; ++m)
          dst[(j*d2 + lq)*nr + ro + m] = (float)(R[j][lq][m] * sc);
  }
  __syncthreads();
  // ---- phase 2: padded B tables for the node WMMA kernel ----
  const int CGB[5] = {1,10,35,116,341};   // cg base per pairtype (sp,sd,pp,pd,dd)
  const int NRM[5] = {3,5,9,15,25};
  const int NTT[5] = {1,1,1,1,2};         // 16-wide N tiles
  const int KPR[5] = {2,4,6,8,14};        // K pairs = 2*ceil(nrme/4)
  const int OFF[5] = {0,64,192,384,640};  // float offset inside Bp region
  float* bp = cgws + BPOFF;
  for (int p = 0; p < 5; ++p) {
    const int nt16 = 16*NTT[p];
    const int len = KPR[p]*nt16*2;
    const float* g = cgws + CGB[p];
    float* d = bp + OFF[p];
    const int nr = NRM[p];
    for (int i = t; i < len; i += 32) {
      int w = i & 1;
      int rest = i >> 1;
      int col = rest % nt16;
      int kp = rest / nt16;
      int k = kp*2 + w;
      d[i] = (k < nr && col < nr) ? g[col*nr + k] : 0.f;
    }
  }
}

// ============================ Edge kernel ================================
__global__ __launch_bounds__(EBLK) void edge_kernel(
    const float* __restrict__ ef, const float* __restrict__ ev,
    const float* __restrict__ cg, float* __restrict__ out, int n_edge)
{
  __shared__ __align__(16) float srow[EBLK * FEAT];
  const int tid = threadIdx.x;
  const long long base  = (long long)blockIdx.x * (EBLK * FEAT);
  const long long total = (long long)n_edge * FEAT;
  long long remll = total - base;
  const int remi  = (int)(remll < (long long)(EBLK*FEAT) ? remll : (long long)(EBLK*FEAT));
  const int rem4  = remi >> 2;
  const float4* gsrc4 = (const float4*)ef + (base >> 2);
  float4*       ldst4 = (float4*)srow;

  // ---- stage 64 feature rows, coalesced b128 ----
#if HAVE_ASYNC
  for (int i = tid; i < rem4; i += EBLK) async_g2l_b128(gsrc4 + i, ldst4 + i);
#else
  for (int i = tid; i < rem4; i += EBLK) ldst4[i] = gsrc4[i];
#endif
  for (int i = rem4*4 + tid; i < remi; i += EBLK) srow[i] = ef[base + i];

  // ---- per-edge Wigner D matrices (closed form, no trig) ----
  const int e = blockIdx.x * EBLK + tid;
  const bool valid = (e < n_edge);
  float D1[3][3], D2[5][5];
  {
    int es = valid ? e : 0;
    float wx = ev[3*es+0], wy = ev[3*es+1], wz = ev[3*es+2];
    float p0 = wy, p1 = wz, p2 = wx;                     // permute [1,2,0]
    float nn = sqrtf(p0*p0 + p1*p1 + p2*p2);
    float inv = 1.0f / fmaxf(nn, 1e-12f);
    float u0 = fminf(fmaxf(p0*inv, -1.f), 1.f);
    float u1 = fminf(fmaxf(p1*inv, -1.f), 1.f);
    float u2 = fminf(fmaxf(p2*inv, -1.f), 1.f);
    float cb = u1;
    float sb = sqrtf(fmaxf(1.f - cb*cb, 0.f));           // sin(arccos)
    float r2 = sqrtf(u0*u0 + u2*u2);
    float ca, sa;
    if (r2 > 1e-20f) { float ir = 1.f / r2; ca = u2*ir; sa = u0*ir; }
    else             { ca = 1.f; sa = 0.f; }
    D1[0][0] = ca;   D1[0][1] = sa*sb; D1[0][2] = sa*cb;
    D1[1][0] = 0.f;  D1[1][1] = cb;    D1[1][2] = -sb;
    D1[2][0] = -sa;  D1[2][1] = ca*sb; D1[2][2] = ca*cb;
    const float SQ3 = 1.7320508075688772f;
    float s2a = 2.f*sa*ca, c2a = ca*ca - sa*sa;
    float s2b = 2.f*sb*cb, c2b = 1.f - 2.f*sb*sb;
    float B22 = 0.25f*(3.f*c2b + 1.f), B23 = -0.5f*SQ3*s2b, B24 = 0.25f*SQ3*(1.f - c2b);
    float B32 = 0.5f*SQ3*s2b, B33 = c2b, B34 = -0.5f*s2b;
    float B42 = B24, B43 = 0.5f*s2b, B44 = 0.25f*(3.f + c2b);
    D2[0][0] = c2a*cb;  D2[0][1] = c2a*sb;  D2[0][2] = s2a*B42; D2[0][3] = s2a*B43; D2[0][4] = s2a*B44;
    D2[1][0] = -ca*sb;  D2[1][1] = ca*cb;   D2[1][2] = sa*B32;  D2[1][3] = sa*B33;  D2[1][4] = sa*B34;
    D2[2][0] = 0.f;     D2[2][1] = 0.f;     D2[2][2] = B22;     D2[2][3] = B23;     D2[2][4] = B24;
    D2[3][0] = sa*sb;   D2[3][1] = -sa*cb;  D2[3][2] = ca*B32;  D2[3][3] = ca*B33;  D2[3][4] = ca*B34;
    D2[4][0] = -s2a*cb; D2[4][1] = -s2a*sb; D2[4][2] = c2a*B42; D2[4][3] = c2a*B43; D2[4][4] = c2a*B44;
  }
#if HAVE_ASYNC
  __builtin_amdgcn_s_wait_asynccnt(0);
#endif
  __syncthreads();

  if (valid) {
    float* row = &srow[tid * FEAT];
    // ---- s-s (D0 = 1): scale by cg ----
    {
      float g0 = cg[0];
      #pragma unroll
      for (int i = 0; i < 9; ++i) row[i] *= g0;
    }
    // ---- s-p ----
    {
      const float* g = cg + 1;                 // [m(3)][r(3)]
      #pragma unroll
      for (int q = 0; q < 6; ++q) {
        float in3[3], hz[3];
        #pragma unroll
        for (int r = 0; r < 3; ++r) in3[r] = row[9 + q*3 + r];
        #pragma unroll
        for (int m = 0; m < 3; ++m) {
          float s = 0.f;
          #pragma unroll
          for (int r = 0; r < 3; ++r) s += g[m*3 + r] * in3[r];
          hz[m] = s;
        }
        #pragma unroll
        for (int k = 0; k < 3; ++k) {
          float s = 0.f;
          #pragma unroll
          for (int o = 0; o < 3; ++o) s += D1[k][o] * hz[o];
          row[9 + q*3 + k] = s;
        }
      }
    }
    // ---- s-d ----
    {
      const float* g = cg + 10;                // [m(5)][r(5)]
      #pragma unroll
      for (int q = 0; q < 3; ++q) {
        float in5[5], hz[5];
        #pragma unroll
        for (int r = 0; r < 5; ++r) in5[r] = row[27 + q*5 + r];
        #pragma unroll
        for (int m = 0; m < 5; ++m) {
          float s = 0.f;
          #pragma unroll
          for (int r = 0; r < 5; ++r) s += g[m*5 + r] * in5[r];
          hz[m] = s;
        }
        #pragma unroll
        for (int k = 0; k < 5; ++k) {
          float s = 0.f;
          #pragma unroll
          for (int o = 0; o < 5; ++o) s += D2[k][o] * hz[o];
          row[27 + q*5 + k] = s;
        }
      }
    }
    // ---- p-p ----
    {
      const float* g = cg + 35;                // [(m*3+o)][r(9)]
      #pragma unroll
      for (int q = 0; q < 4; ++q) {
        float in9[9], Hz[3][3], T[3][3];
        #pragma unroll
        for (int r = 0; r < 9; ++r) in9[r] = row[42 + q*9 + r];
        #pragma unroll
        for (int m = 0; m < 3; ++m)
          #pragma unroll
          for (int o = 0; o < 3; ++o) {
            float s = 0.f;
            #pragma unroll
            for (int r = 0; r < 9; ++r) s += g[(m*3 + o)*9 + r] * in9[r];
            Hz[m][o] = s;
          }
        #pragma unroll
        for (int l = 0; l < 3; ++l)
          #pragma unroll
          for (int o = 0; o < 3; ++o) {
            float s = 0.f;
            #pragma unroll
            for (int m = 0; m < 3; ++m) s += D1[l][m] * Hz[m][o];
            T[l][o] = s;
          }
        #pragma unroll
        for (int l = 0; l < 3; ++l)
          #pragma unroll
          for (int k = 0; k < 3; ++k) {
            float s = 0.f;
            #pragma unroll
            for (int o = 0; o < 3; ++o) s += T[l][o] * D1[k][o];
            row[42 + q*9 + l*3 + k] = s;
          }
      }
    }
    // ---- p-d ----
    {
      const float* g = cg + 116;               // [(m*5+o)][r(15)]
      #pragma unroll
      for (int q = 0; q < 2; ++q) {
        float in15[15], Hz[3][5], T[3][5];
        #pragma unroll
        for (int r = 0; r < 15; ++r) in15[r] = row[78 + q*15 + r];
        #pragma unroll
        for (int m = 0; m < 3; ++m)
          #pragma unroll
          for (int o = 0; o < 5; ++o) {
            float s = 0.f;
            #pragma unroll
            for (int r = 0; r < 15; ++r) s += g[(m*5 + o)*15 + r] * in15[r];
            Hz[m][o] = s;
          }
        #pragma unroll
        for (int l = 0; l < 3; ++l)
          #pragma unroll
          for (int o = 0; o < 5; ++o) {
            float s = 0.f;
            #pragma unroll
            for (int m = 0; m < 3; ++m) s += D1[l][m] * Hz[m][o];
            T[l][o] = s;
          }
        #pragma unroll
        for (int l = 0; l < 3; ++l)
          #pragma unroll
          for (int k = 0; k < 5; ++k) {
            float s = 0.f;
            #pragma unroll
            for (int o = 0; o < 5; ++o) s += T[l][o] * D2[k][o];
            row[78 + q*15 + l*5 + k] = s;
          }
      }
    }
    // ---- d-d ----
    {
      const float* g = cg + 341;               // [(m*5+o)][r(25)]
      float in25[25], Hz[5][5], T[5][5];
      #pragma unroll
      for (int r = 0; r < 25; ++r) in25[r] = row[108 + r];
      #pragma unroll
      for (int m = 0; m < 5; ++m)
        #pragma unroll
        for (int o = 0; o < 5; ++o) {
          float s = 0.f;
          #pragma unroll
          for (int r = 0; r < 25; ++r) s += g[(m*5 + o)*25 + r] * in25[r];
          Hz[m][o] = s;
        }
      #pragma unroll
      for (int l = 0; l < 5; ++l)
        #pragma unroll
        for (int o = 0; o < 5; ++o) {
          float s = 0.f;
          #pragma unroll
          for (int m = 0; m < 5; ++m) s += D2[l][m] * Hz[m][o];
          T[l][o] = s;
        }
      #pragma unroll
      for (int l = 0; l < 5; ++l)
        #pragma unroll
        for (int k = 0; k < 5; ++k) {
          float s = 0.f;
          #pragma unroll
          for (int o = 0; o < 5; ++o) s += T[l][o] * D2[k][o];
          row[108 + l*5 + k] = s;
        }
    }
  }
  __syncthreads();

  // ---- store back, coalesced b128 ----
  float4* gdst4 = (float4*)out + (base >> 2);
#if HAVE_ASYNC
  for (int i = tid; i < rem4; i += EBLK) async_l2g_b128(gdst4 + i, ldst4 + i);
  // S_ENDPGM implicitly waits for all counters
#else
  for (int i = tid; i < rem4; i += EBLK) gdst4[i] = ldst4[i];
#endif
  for (int i = rem4*4 + tid; i < remi; i += EBLK) out[base + i] = srow[i];
}

// ============================ Node kernel (WMMA) =========================
// out[n, q, lm] = sum_r A[n,q,r] * B[r,lm].  M-tile = 16 consecutive nodes
// at fixed q (no div/mod), B from zero-padded LDS tiles (unconditional b64).
template<int NPAIR, int NRME>
__device__ __forceinline__ void node_block(
    const float* __restrict__ nf, float* __restrict__ out,
    const float* sbp, int bpoff, int start, int nbase, int lane, int n_node)
{
  constexpr int NT = (NRME + 15) / 16;
  constexpr int KC = (NRME + 3) / 4;
  const int m = lane & 15, hi = lane >> 4;
  const bool full = (nbase + 16 <= n_node);         // wave-uniform fast path
  int nA = nbase + m;
  if (nA >= n_node) nA = n_node - 1;                // clamp loads only
  #pragma unroll
  for (int q = 0; q < NPAIR; ++q) {
    const float* arow = nf + (size_t)nA*FEAT + start + q*NRME;
    #pragma unroll
    for (int nt = 0; nt < NT; ++nt) {
      v8f c = {0.f,0.f,0.f,0.f,0.f,0.f,0.f,0.f};
      #pragma unroll
      for (int kc = 0; kc < KC; ++kc) {
        const int k0 = kc*4 + 2*hi;
        const int i0 = (k0     < NRME) ? k0     : 0;
        const int i1 = (k0 + 1 < NRME) ? k0 + 1 : 0;
        float ax = arow[i0];
        float ay = arow[i1];
        v2f a;
        a.x = (k0     < NRME) ? ax : 0.f;
        a.y = (k0 + 1 < NRME) ? ay : 0.f;
        v2f b = *(const v2f*)(sbp + bpoff + (((kc*2 + hi)*(16*NT)) + nt*16 + m)*2);
        c = __builtin_amdgcn_wmma_f32_16x16x4_f32(false, a, false, b, (short)0, c, false, false);
      }
      const int col = nt*16 + m;
      if (col < NRME) {
        float* orow = out + (size_t)(nbase + 8*hi)*FEAT + start + q*NRME + col;
        if (full) {
          #pragma unroll
          for (int r = 0; r < 8; ++r) orow[(size_t)r*FEAT] = c[r];
        } else {
          #pragma unroll
          for (int r = 0; r < 8; ++r)
            if (nbase + 8*hi + r < n_node)          // C row r+8*hi -> node
              orow[(size_t)r*FEAT] = c[r];
        }
      }
    }
  }
}

__global__ __launch_bounds__(128) void node_kernel(
    const float* __restrict__ nf, const float* __restrict__ cgws,
    float* __restrict__ out, int n_node)
{
  __shared__ __align__(16) float sbp[1536];
  const float* bpg = cgws + BPOFF;
#if HAVE_ASYNC
  for (int i = threadIdx.x; i < 384; i += 128)
    async_g2l_b128((const float4*)bpg + i, (float4*)sbp + i);
  __builtin_amdgcn_s_wait_asynccnt(0);
#else
  for (int i = threadIdx.x; i < 384; i += 128)
    ((float4*)sbp)[i] = ((const float4*)bpg)[i];
#endif
  __syncthreads();
  const int wave = threadIdx.x >> 5, lane = threadIdx.x & 31;
  const int wtile = blockIdx.x * 4 + wave;
  const int ntiles = (n_node + 15) / 16;
  if (wtile >= ntiles) return;                       // wave-uniform
  const int nbase = wtile * 16;

  // s-s: plain scaled copy (uniform scalar load of cg[0])
  {
    float g0 = cgws[0];
    if (nbase + 16 <= n_node) {
      for (int idx = lane; idx < 16*9; idx += 32) {
        int n = nbase + idx/9, i = idx % 9;
        out[(size_t)n*FEAT + i] = g0 * nf[(size_t)n*FEAT + i];
      }
    } else {
      for (int idx = lane; idx < 16*9; idx += 32) {
        int n = nbase + idx/9, i = idx % 9;
        if (n < n_node) out[(size_t)n*FEAT + i] = g0 * nf[(size_t)n*FEAT + i];
      }
    }
  }
  node_block<6, 3 >(nf, out, sbp, 0,   9,   nbase, lane, n_node);
  node_block<3, 5 >(nf, out, sbp, 64,  27,  nbase, lane, n_node);
  node_block<4, 9 >(nf, out, sbp, 192, 42,  nbase, lane, n_node);
  node_block<2, 15>(nf, out, sbp, 384, 78,  nbase, lane, n_node);
  node_block<1, 25>(nf, out, sbp, 640, 108, nbase, lane, n_node);
}

// =============================== launch ==================================
extern "C" void kernel_launch(void* const* d_in, const int* in_sizes, int n_in,
                              void* d_out, int out_size, void* d_ws, size_t ws_size,
                              hipStream_t stream) {
  const float* ef  = (const float*)d_in[0];
  const float* nf  = (const float*)d_in[1];
  const float* ev  = (const float*)d_in[2];
  // d_in[3] = edge_index: unused by the reference forward
  const int n_edge = in_sizes[0] / FEAT;
  const int n_node = in_sizes[1] / FEAT;
  float* cgws = (float*)d_ws;
  float* out  = (float*)d_out;

  init_cg_kernel<<<1, 32, 0, stream>>>(cgws);

  const int eblocks = (n_edge + EBLK - 1) / EBLK;
  edge_kernel<<<eblocks, EBLK, 0, stream>>>(ef, ev, cgws, out, n_edge);

  const int wtiles = (n_node + 15) / 16;
  const int nblocks = (wtiles + 3) / 4;
  node_kernel<<<nblocks, 128, 0, stream>>>(nf, cgws, out + (size_t)n_edge * FEAT, n_node);
}